// neighbor_proj_loss_53601191854658
// MI455X (gfx1250) — compile-verified
//
#include <hip/hip_runtime.h>
#include <hip/hip_bf16.h>
#include <math.h>

// ---------------- problem constants (match reference) ----------------
#define PP   256   // proxies
#define NN   256   // points
#define DD   512   // embedding dim
#define KK   8     // plane rank
#define QQ   (PP*KK)          // 2048 flattened proxy-plane rows
#define NK   (NN*KK)          // 2048 flattened teacher rows
#define DELTA_C   1.0f
#define INV142    (1.0f/1.42f)
#define EPSN      1e-12f

typedef __attribute__((ext_vector_type(2))) float v2f;
typedef __attribute__((ext_vector_type(8))) float v8f;

// ---------------- workspace layout (floats) ----------------
// all offsets are compile-time; total ~27.2MB
#define OFF_T     ((size_t)0)                         // NN*DD    normalized t
#define OFF_PROX  (OFF_T    + (size_t)NN*DD)          // PP*DD    normalized proxies_f
#define OFF_PPN   (OFF_PROX + (size_t)PP*DD)          // QQ*DD    row-normalized proxy_planes_f
#define OFF_S2    (OFF_PPN  + (size_t)QQ*DD)          // NN       ||s_n||^2
#define OFF_ST    (OFF_S2   + (size_t)NN)             // NN*KK    s_n . teacher_nk
#define OFF_TT    (OFF_ST   + (size_t)NN*KK)          // NN*KK    t_n . teacher_nk
#define OFF_PTP   (OFF_TT   + (size_t)NN*KK)          // PP*KK    prox_p . plane_pk
#define OFF_GT    (OFF_PTP  + (size_t)PP*KK)          // NN*64    teacher Gram per point
#define OFF_GP    (OFF_GT   + (size_t)NN*64)          // PP*64    plane Gram per proxy
#define OFF_W     (OFF_GP   + (size_t)PP*64)          // PP*64    masked U/s transform
#define OFF_TP    (OFF_W    + (size_t)PP*64)          // NN*QQ    Tp[n, p*8+k]
#define OFF_PT    (OFF_TP   + (size_t)NN*QQ)          // PP*NK    Pt[p, n*8+k]
#define OFF_PS    (OFF_PT   + (size_t)PP*NK)          // PP*NN    prox . s
#define OFF_TPR   (OFF_PS   + (size_t)PP*NN)          // PP*NN    prox . t
#define OFF_SIM   (OFF_TPR  + (size_t)PP*NN)          // PP*NN    sim + t_sim^T
#define OFF_MC    (OFF_SIM  + (size_t)PP*NN)          // QQ*NK    Mcos[q, n*8+k]
#define OFF_L1    (OFF_MC   + (size_t)QQ*NK)          // 256      loss1 block partials
#define OFF_L2    (OFF_L1   + (size_t)256)            // 2048     loss2 block partials

// ---------------- helpers ----------------
__device__ __forceinline__ float warp_sum(float v) {
  #pragma unroll
  for (int off = 16; off; off >>= 1) v += __shfl_xor(v, off, 32);
  return v;
}

// block (256 threads) sum; valid on thread 0
__device__ __forceinline__ float block_sum(float v) {
  __shared__ float red[8];
  v = warp_sum(v);
  if ((threadIdx.x & 31) == 0) red[threadIdx.x >> 5] = v;
  __syncthreads();
  float tot = 0.0f;
  if (threadIdx.x == 0) {
    #pragma unroll
    for (int i = 0; i < 8; ++i) tot += red[i];
  }
  __syncthreads();
  return tot;
}

// ---------------- row normalize / norms ----------------
// grid.x = rows, 256 threads. dst may be null (norms only), n2out may be null.
__global__ __launch_bounds__(256) void rownorm_kernel(const float* __restrict__ src,
                                                      float* __restrict__ dst,
                                                      float* __restrict__ n2out,
                                                      int cols) {
  const size_t row = blockIdx.x;
  const float* s = src + row * (size_t)cols;
  float acc = 0.0f;
  for (int c = threadIdx.x; c < cols; c += 256) { float v = s[c]; acc += v * v; }
  float tot = block_sum(acc);
  __shared__ float sc_sh;
  if (threadIdx.x == 0) {
    if (n2out) n2out[row] = tot;
    sc_sh = 1.0f / fmaxf(sqrtf(tot), EPSN);
  }
  __syncthreads();
  if (dst) {
    float sc = sc_sh;
    float* d = dst + row * (size_t)cols;
    for (int c = threadIdx.x; c < cols; c += 256) d[c] = s[c] * sc;
  }
}

// ---------------- per-point small dots: st, tt, Gt ----------------
// grid.x = NN, 256 threads = 8 waves; 80 dot-products of length DD per point
__global__ __launch_bounds__(256) void point_dots_kernel(const float* __restrict__ s_emb,
                                                         const float* __restrict__ t,
                                                         const float* __restrict__ teacher,
                                                         float* __restrict__ st,
                                                         float* __restrict__ tt,
                                                         float* __restrict__ Gt) {
  const int n = blockIdx.x;
  const float* T = teacher + (size_t)n * KK * DD;
  const int wave = threadIdx.x >> 5, lane = threadIdx.x & 31;
  for (int j = wave; j < 80; j += 8) {
    const float* a; const float* b;
    if (j < 8)       { a = s_emb + (size_t)n * DD; b = T + (size_t)j * DD; }
    else if (j < 16) { a = t     + (size_t)n * DD; b = T + (size_t)(j - 8) * DD; }
    else             { int k = (j - 16) >> 3, k0 = (j - 16) & 7;
                       a = T + (size_t)k * DD;     b = T + (size_t)k0 * DD; }
    float acc = 0.0f;
    for (int d = lane; d < DD; d += 32) acc += a[d] * b[d];
    acc = warp_sum(acc);
    if (lane == 0) {
      if (j < 8)       st[(size_t)n * KK + j] = acc;
      else if (j < 16) tt[(size_t)n * KK + (j - 8)] = acc;
      else             Gt[(size_t)n * 64 + (j - 16)] = acc;
    }
  }
}

// ---------------- per-proxy small dots: proxTp, Gp ----------------
__global__ __launch_bounds__(256) void proxy_dots_kernel(const float* __restrict__ prox,
                                                         const float* __restrict__ planes,
                                                         float* __restrict__ pTp,
                                                         float* __restrict__ Gp) {
  const int p = blockIdx.x;
  const float* PL = planes + (size_t)p * KK * DD;
  const int wave = threadIdx.x >> 5, lane = threadIdx.x & 31;
  for (int j = wave; j < 72; j += 8) {
    const float* a; const float* b;
    if (j < 8) { a = prox + (size_t)p * DD; b = PL + (size_t)j * DD; }
    else       { int k = (j - 8) >> 3, k0 = (j - 8) & 7;
                 a = PL + (size_t)k * DD;   b = PL + (size_t)k0 * DD; }
    float acc = 0.0f;
    for (int d = lane; d < DD; d += 32) acc += a[d] * b[d];
    acc = warp_sum(acc);
    if (lane == 0) {
      if (j < 8) pTp[(size_t)p * KK + j] = acc;
      else       Gp[(size_t)p * 64 + (j - 8)] = acc;
    }
  }
}

// ---------------- per-proxy 8x8 Jacobi eigendecomposition ----------------
// Produces W[p][k][k0] = mask_k * U[k0][ord_k] / s_k  (s_k = singular values desc)
__global__ __launch_bounds__(256) void eigen_kernel(const float* __restrict__ Gp,
                                                    float* __restrict__ W) {
  const int p = blockIdx.x * blockDim.x + threadIdx.x;
  if (p >= PP) return;
  float A[8][8], V[8][8];
  #pragma unroll
  for (int i = 0; i < 8; ++i)
    #pragma unroll
    for (int j = 0; j < 8; ++j) {
      A[i][j] = Gp[(size_t)p * 64 + i * 8 + j];
      V[i][j] = (i == j) ? 1.0f : 0.0f;
    }
  for (int sweep = 0; sweep < 12; ++sweep) {
    for (int i = 0; i < 7; ++i) {
      for (int j = i + 1; j < 8; ++j) {
        float apq = A[i][j];
        if (fabsf(apq) < 1e-30f) continue;
        float theta = (A[j][j] - A[i][i]) / (2.0f * apq);
        float tv = (theta >= 0.0f ? 1.0f : -1.0f) / (fabsf(theta) + sqrtf(theta * theta + 1.0f));
        float c = rsqrtf(1.0f + tv * tv);
        float s = tv * c;
        for (int k = 0; k < 8; ++k) { float aik = A[i][k], ajk = A[j][k];
          A[i][k] = c * aik - s * ajk; A[j][k] = s * aik + c * ajk; }
        for (int k = 0; k < 8; ++k) { float aki = A[k][i], akj = A[k][j];
          A[k][i] = c * aki - s * akj; A[k][j] = s * aki + c * akj; }
        for (int k = 0; k < 8; ++k) { float vki = V[k][i], vkj = V[k][j];
          V[k][i] = c * vki - s * vkj; V[k][j] = s * vki + c * vkj; }
      }
    }
  }
  float ev[8]; int ord[8];
  #pragma unroll
  for (int k = 0; k < 8; ++k) { ev[k] = A[k][k]; ord[k] = k; }
  for (int a = 1; a < 8; ++a) {                  // insertion sort, descending
    float e = ev[a]; int o = ord[a]; int b = a - 1;
    while (b >= 0 && ev[b] < e) { ev[b + 1] = ev[b]; ord[b + 1] = ord[b]; --b; }
    ev[b + 1] = e; ord[b + 1] = o;
  }
  float sv[8], tot = 0.0f;
  #pragma unroll
  for (int k = 0; k < 8; ++k) { sv[k] = sqrtf(fmaxf(ev[k], 0.0f)); tot += sv[k]; }
  float inv_tot = (tot > 1e-20f) ? 1.0f / tot : 0.0f;
  int rank = 7; float cum = 0.0f;
  for (int k = 0; k < 8; ++k) { cum += sv[k]; if (cum * inv_tot >= 0.99f) { rank = k; break; } }
  for (int k = 0; k < 8; ++k) {
    float inv = (k <= rank) ? 1.0f / fmaxf(sv[k], EPSN) : 0.0f;
    for (int k0 = 0; k0 < 8; ++k0)
      W[(size_t)p * 64 + k * 8 + k0] = inv * V[k0][ord[k]];
  }
}

// ---------------- generic C = A * B^T via V_WMMA_F32_16X16X4_F32 ----------------
// A: M x D row-major, B: N x D row-major, C: M x N row-major.
// Block tile 128(M) x 64(N), 8 waves, each wave 32x32 (2x2 WMMA accumulators).
#define BM 128
#define BN 64
#define BK 32
#define LDSP 40   // padded LDS row stride (floats): 160B = 16B-aligned float4 rows

__global__ __launch_bounds__(256) void gemm_nt_wmma(const float* __restrict__ A,
                                                    const float* __restrict__ B,
                                                    float* __restrict__ C,
                                                    int M, int Nc, int Dd) {
  __shared__ float As[BM][LDSP];
  __shared__ float Bs[BN][LDSP];
  const int tid  = threadIdx.x;
  const int wave = tid >> 5;
  const int lane = tid & 31;
  const int half = lane >> 4;      // K-pair selector (WMMA 32-bit A/B layout)
  const int r    = lane & 15;      // row/col within 16
  const int bm = blockIdx.y * BM;
  const int bn = blockIdx.x * BN;
  const int wm = (wave >> 1) * 32; // 4 waves along M
  const int wn = (wave & 1) * 32;  // 2 waves along N

  v8f acc[2][2] = {};

  for (int kt = 0; kt < Dd; kt += BK) {
    // stage A tile: 128x32 floats = 1024 float4, 4 per thread
    #pragma unroll
    for (int i = 0; i < 4; ++i) {
      int v = tid + i * 256;
      int row = v >> 3;            // 8 float4 per row
      int col = (v & 7) * 4;
      const float* gp = A + (size_t)(bm + row) * Dd + kt + col;
      float4 d = *(const float4*)gp;
      *(float4*)(&As[row][col]) = d;
      if (kt + BK < Dd) __builtin_prefetch(gp + BK, 0, 1);   // global_prefetch next K-tile
    }
    // stage B tile: 64x32 floats = 512 float4, 2 per thread
    #pragma unroll
    for (int i = 0; i < 2; ++i) {
      int v = tid + i * 256;
      int row = v >> 3;
      int col = (v & 7) * 4;
      const float* gp = B + (size_t)(bn + row) * Dd + kt + col;
      float4 d = *(const float4*)gp;
      *(float4*)(&Bs[row][col]) = d;
      if (kt + BK < Dd) __builtin_prefetch(gp + BK, 0, 1);
    }
    __syncthreads();

    #pragma unroll
    for (int kk = 0; kk < BK; kk += 4) {
#if __has_builtin(__builtin_amdgcn_wmma_f32_16x16x4_f32)
      v2f a[2], b[2];
      #pragma unroll
      for (int i = 0; i < 2; ++i) {
        // 32-bit A 16x4 layout: lanes 0-15 -> K=0,1 ; lanes 16-31 -> K=2,3
        a[i].x = As[wm + i * 16 + r][kk + 2 * half + 0];
        a[i].y = As[wm + i * 16 + r][kk + 2 * half + 1];
        b[i].x = Bs[wn + i * 16 + r][kk + 2 * half + 0];
        b[i].y = Bs[wn + i * 16 + r][kk + 2 * half + 1];
      }
      #pragma unroll
      for (int i = 0; i < 2; ++i)
        #pragma unroll
        for (int j = 0; j < 2; ++j)
          acc[i][j] = __builtin_amdgcn_wmma_f32_16x16x4_f32(
              false, a[i], false, b[j], (short)0, acc[i][j], false, false);
#else
      // scalar fallback mirroring the WMMA C/D lane layout
      #pragma unroll
      for (int i = 0; i < 2; ++i)
        #pragma unroll
        for (int j = 0; j < 2; ++j)
          #pragma unroll
          for (int v = 0; v < 8; ++v) {
            int mrow = wm + i * 16 + v + 8 * half;
            int ncol = wn + j * 16 + r;
            float s = 0.0f;
            #pragma unroll
            for (int k4 = 0; k4 < 4; ++k4) s += As[mrow][kk + k4] * Bs[ncol][kk + k4];
            acc[i][j][v] += s;
          }
#endif
    }
    __syncthreads();
  }

  // store: D layout — VGPR v: lanes 0-15 -> M=v, lanes 16-31 -> M=v+8; N=r
  #pragma unroll
  for (int i = 0; i < 2; ++i)
    #pragma unroll
    for (int j = 0; j < 2; ++j)
      #pragma unroll
      for (int v = 0; v < 8; ++v)
        C[(size_t)(bm + wm + i * 16 + v + 8 * half) * Nc + (bn + wn + j * 16 + r)] = acc[i][j][v];
}

// ---------------- fused similarity + loss1 epilogue ----------------
// One block per proxy p (256 threads = 256 n values).
__global__ __launch_bounds__(256) void sim_loss1_kernel(const float* __restrict__ Tp,
                                                        const float* __restrict__ pTp,
                                                        const float* __restrict__ W,
                                                        const float* __restrict__ Pt,
                                                        const float* __restrict__ st,
                                                        const float* __restrict__ tt,
                                                        const float* __restrict__ Gt,
                                                        const float* __restrict__ s2,
                                                        const float* __restrict__ ps,
                                                        const float* __restrict__ tpr,
                                                        float* __restrict__ simbuf,
                                                        float* __restrict__ l1p) {
  const int p = blockIdx.x;
  const int n = threadIdx.x;
  __shared__ float Wsh[64];
  __shared__ float pTpsh[8];
  if (threadIdx.x < 64) Wsh[threadIdx.x] = W[(size_t)p * 64 + threadIdx.x];
  if (threadIdx.x < 8)  pTpsh[threadIdx.x] = pTp[(size_t)p * KK + threadIdx.x];
  __syncthreads();

  // ---- forward part: residues + proj (both from pc) ----
  float pc[8]; float proj2 = 0.0f;
  #pragma unroll
  for (int k = 0; k < 8; ++k) {
    pc[k] = Tp[(size_t)n * QQ + p * KK + k] - pTpsh[k];
    proj2 += pc[k] * pc[k];
  }
  float dot2 = 0.0f;
  #pragma unroll
  for (int k = 0; k < 8; ++k) {
    float w = 0.0f;
    #pragma unroll
    for (int k0 = 0; k0 < 8; ++k0) w += pc[k0] * Wsh[k * 8 + k0];
    dot2 += w * w;
  }
  float ctt = 2.0f - 2.0f * tpr[(size_t)p * NN + n];           // ||t-prox||^2
  float res = sqrtf(fmaxf(ctt - dot2, 0.0f));
  float proj = sqrtf(proj2);
  float ra = 1.0f / (1.0f + res * INV142);
  float rb = 1.0f / (1.0f + proj);
  float sim = (ra * ra) * (rb * rb);

  // ---- transpose part: t_res (uses s_emb) + t_proj (uses t) ----
  float q[8]; float q2 = 0.0f, tc2 = 0.0f;
  #pragma unroll
  for (int k = 0; k < 8; ++k) {
    float ptk = Pt[(size_t)p * NK + n * KK + k];
    q[k] = ptk - st[(size_t)n * KK + k];
    q2 += q[k] * q[k];
    float tck = ptk - tt[(size_t)n * KK + k];
    tc2 += tck * tck;
  }
  float qGq = 0.0f;
  #pragma unroll
  for (int k = 0; k < 8; ++k) {
    float a = 0.0f;
    #pragma unroll
    for (int k0 = 0; k0 < 8; ++k0) a += Gt[(size_t)n * 64 + k * 8 + k0] * q[k0];
    qGq += q[k] * a;
  }
  float d2 = 1.0f + s2[n] - 2.0f * ps[(size_t)p * NN + n];     // ||prox-s||^2
  float tres = sqrtf(fmaxf(d2 - 2.0f * q2 + qGq, 0.0f));
  float tproj = sqrtf(tc2);
  float ta = 1.0f / (1.0f + tres * INV142);
  float tb = 1.0f / (1.0f + tproj);
  float tsim = (ta * ta) * (tb * tb);

  float simT = sim + tsim;
  simbuf[(size_t)p * NN + n] = simT;

  float Sd = sqrtf(fmaxf(d2, 0.0f));
  float diff = DELTA_C * (1.0f - simT) - Sd;
  float tot = block_sum(diff * diff);
  if (threadIdx.x == 0) l1p[blockIdx.x] = tot;
}

// ---------------- loss2 epilogue ----------------
// One block per flattened plane row q (2048 blocks, 256 threads = n values)
__global__ __launch_bounds__(256) void loss2_kernel(const float* __restrict__ Mc,
                                                    const float* __restrict__ simbuf,
                                                    float* __restrict__ l2p) {
  const int qrow = blockIdx.x;          // 0..QQ-1
  const int n = threadIdx.x;
  float c2 = 0.0f;
  #pragma unroll
  for (int k = 0; k < 8; ++k) {
    float m = Mc[(size_t)qrow * NK + n * KK + k];
    c2 += m * m;
  }
  float cosv = sqrtf(c2);
  float diff = simbuf[(size_t)(qrow >> 3) * NN + n] - cosv;
  float tot = block_sum(diff * diff);
  if (threadIdx.x == 0) l2p[blockIdx.x] = tot;
}

// ---------------- deterministic final reduction ----------------
__global__ __launch_bounds__(256) void final_reduce_kernel(const float* __restrict__ l1p,
                                                           const float* __restrict__ l2p,
                                                           float* __restrict__ out) {
  float a = (threadIdx.x < 256) ? l1p[threadIdx.x] : 0.0f;
  float b = 0.0f;
  for (int i = threadIdx.x; i < 2048; i += 256) b += l2p[i];
  float v = a * (1.0f / ((float)PP * NN)) + b * (1.0f / ((float)QQ * NN));
  float tot = block_sum(v);
  if (threadIdx.x == 0) out[0] = tot;
}

// ---------------- host launch ----------------
extern "C" void kernel_launch(void* const* d_in, const int* in_sizes, int n_in,
                              void* d_out, int out_size, void* d_ws, size_t ws_size,
                              hipStream_t stream) {
  (void)in_sizes; (void)n_in; (void)out_size; (void)ws_size;
  const float* s_emb   = (const float*)d_in[0];   // (N, D)
  const float* t_emb   = (const float*)d_in[1];   // (N, D)
  const float* teacher = (const float*)d_in[2];   // (N, K, D)
  const float* prox_f  = (const float*)d_in[3];   // (P, D)
  // d_in[4] proxies_g, d_in[5] proxy_planes_g, d_in[7] epoch: unused by reference path
  const float* planes  = (const float*)d_in[6];   // (P, K, D)
  float* out = (float*)d_out;
  float* ws  = (float*)d_ws;

  float* t    = ws + OFF_T;    float* prox = ws + OFF_PROX; float* ppn = ws + OFF_PPN;
  float* s2   = ws + OFF_S2;   float* st   = ws + OFF_ST;   float* tt  = ws + OFF_TT;
  float* pTp  = ws + OFF_PTP;  float* Gt   = ws + OFF_GT;   float* Gp  = ws + OFF_GP;
  float* W    = ws + OFF_W;    float* Tp   = ws + OFF_TP;   float* Pt  = ws + OFF_PT;
  float* ps   = ws + OFF_PS;   float* tpr  = ws + OFF_TPR;  float* simb= ws + OFF_SIM;
  float* Mc   = ws + OFF_MC;   float* l1p  = ws + OFF_L1;   float* l2p = ws + OFF_L2;

  // prep: normalizations + norms
  rownorm_kernel<<<NN, 256, 0, stream>>>(t_emb,  t,    nullptr, DD);
  rownorm_kernel<<<PP, 256, 0, stream>>>(prox_f, prox, nullptr, DD);
  rownorm_kernel<<<QQ, 256, 0, stream>>>(planes, ppn,  nullptr, DD);
  rownorm_kernel<<<NN, 256, 0, stream>>>(s_emb,  nullptr, s2,   DD);

  // small per-row dot products / Grams
  point_dots_kernel<<<NN, 256, 0, stream>>>(s_emb, t, teacher, st, tt, Gt);
  proxy_dots_kernel<<<PP, 256, 0, stream>>>(prox, planes, pTp, Gp);
  eigen_kernel<<<1, 256, 0, stream>>>(Gp, W);

  // big GEMMs on the fp32 WMMA pipe
  gemm_nt_wmma<<<dim3(QQ / BN, NN / BM), 256, 0, stream>>>(t,    planes,  Tp, NN, QQ, DD);
  gemm_nt_wmma<<<dim3(NK / BN, PP / BM), 256, 0, stream>>>(prox, teacher, Pt, PP, NK, DD);
  gemm_nt_wmma<<<dim3(NK / BN, QQ / BM), 256, 0, stream>>>(ppn,  teacher, Mc, QQ, NK, DD);
  gemm_nt_wmma<<<dim3(NN / BN, PP / BM), 256, 0, stream>>>(prox, s_emb,   ps, PP, NN, DD);
  gemm_nt_wmma<<<dim3(NN / BN, PP / BM), 256, 0, stream>>>(prox, t,       tpr, PP, NN, DD);

  // epilogues + deterministic reduction
  sim_loss1_kernel<<<PP, 256, 0, stream>>>(Tp, pTp, W, Pt, st, tt, Gt, s2, ps, tpr, simb, l1p);
  loss2_kernel<<<QQ, 256, 0, stream>>>(Mc, simb, l2p);
  final_reduce_kernel<<<1, 256, 0, stream>>>(l1p, l2p, out);
}